// FlashAttentionWithKVCache_1855425872541
// MI455X (gfx1250) — compile-verified
//
#include <hip/hip_runtime.h>
#include <math.h>

typedef float v2f __attribute__((ext_vector_type(2)));
typedef float v8f __attribute__((ext_vector_type(8)));

#define B_      16
#define QL_     32
#define H_      32
#define HKV_    8
#define D_      128
#define MAXSEQ_ 8192
#define REP_    4
#define SCALE_  0.08838834764831845f  /* 1/sqrt(128) */

#define KSTRIDE 132   /* LDS floats per K row (16 keys x 128d), padded */
#define VSTRIDE 18    /* LDS floats per V^T row (128d x 16 keys), padded */
#define PSTRIDE 18    /* LDS floats per P row, padded */
#define NEG_BIG (-1.0e30f)
#define NROWS   128                     /* rep*QL rows per (b,kvh) */
#define WSROW   ((size_t)NROWS * D_ + 2 * NROWS)  /* floats per (b,kvh,split) region */

__device__ __forceinline__ const float* kv_src(const float* __restrict__ fresh,
                                               const float* __restrict__ cache,
                                               int b, int kvh, int j, int cache_len, int d4) {
  if (j < cache_len)
    return cache + ((((size_t)b * MAXSEQ_) + j) * HKV_ + kvh) * (size_t)D_ + d4;
  return fresh + ((((size_t)b * QL_) + (j - cache_len)) * HKV_ + kvh) * (size_t)D_ + d4;
}

template <int SPLIT>
__global__ __launch_bounds__(256, 1)
void fa_kvcache_f32_wmma(const float* __restrict__ q,
                         const float* __restrict__ k,
                         const float* __restrict__ v,
                         const float* __restrict__ k_cache,
                         const float* __restrict__ v_cache,
                         const int* __restrict__ cache_len_p,
                         float* __restrict__ out,
                         float* __restrict__ ws) {
  __shared__ float Ks[2][16 * KSTRIDE];
  __shared__ float VT[2][D_ * VSTRIDE];
  __shared__ float Ps[8 * 16 * PSTRIDE];

  const int cache_len = cache_len_p[0];
  const int kv_len    = cache_len + QL_;

  int t = blockIdx.x;
  const int sid = t % SPLIT;  t /= SPLIT;
  const int kvh = t % HKV_;
  const int b   = t / HKV_;

  /* this split's key range: chunk is a multiple of 16 */
  const int chunk = (((kv_len + SPLIT * 16 - 1) / (SPLIT * 16)) << 4);
  const int start = sid * chunk;
  const int end   = (start + chunk < kv_len) ? (start + chunk) : kv_len;
  const int nb    = (end > start) ? ((end - start + 15) >> 4) : 0;

  const int tid  = threadIdx.x;
  const int wave = tid >> 5;
  const int lane = tid & 31;
  const int lo   = lane & 15;
  const int hi   = lane >> 4;

  const int key0 = tid >> 5;
  const int d40  = (tid & 31) << 2;
  const int key1 = (tid + 256) >> 5;

  /* ---- Q tile as A-matrix fragments, pre-scaled ---- */
  const int r  = wave * 16 + lo;
  const int ql = r & (QL_ - 1);
  const int h  = kvh * REP_ + (r >> 5);
  const float* qrow = q + (((size_t)b * QL_ + ql) * H_ + h) * (size_t)D_;
  v2f qf[32];
#pragma unroll
  for (int c = 0; c < 32; ++c) {
    v2f tq = *(const v2f*)(qrow + 4 * c + 2 * hi);
    qf[c].x = tq.x * SCALE_;
    qf[c].y = tq.y * SCALE_;
  }

  v8f acc[8];
  float mrow[8], lrow[8];
#pragma unroll
  for (int dt = 0; dt < 8; ++dt) {
    v8f z = {0.f, 0.f, 0.f, 0.f, 0.f, 0.f, 0.f, 0.f};
    acc[dt] = z;
  }
#pragma unroll
  for (int i = 0; i < 8; ++i) { mrow[i] = NEG_BIG; lrow[i] = 0.f; }

  /* ---- prologue: stage first block into buffer 0 ---- */
  {
    int j0 = start + key0, j1 = start + key1;
    if (j0 > kv_len - 1) j0 = kv_len - 1;
    if (j1 > kv_len - 1) j1 = kv_len - 1;
    float4 k0 = *(const float4*)kv_src(k, k_cache, b, kvh, j0, cache_len, d40);
    float4 k1 = *(const float4*)kv_src(k, k_cache, b, kvh, j1, cache_len, d40);
    float4 v0 = *(const float4*)kv_src(v, v_cache, b, kvh, j0, cache_len, d40);
    float4 v1 = *(const float4*)kv_src(v, v_cache, b, kvh, j1, cache_len, d40);
    *(float4*)&Ks[0][key0 * KSTRIDE + d40] = k0;
    *(float4*)&Ks[0][key1 * KSTRIDE + d40] = k1;
    VT[0][(d40 + 0) * VSTRIDE + key0] = v0.x;
    VT[0][(d40 + 1) * VSTRIDE + key0] = v0.y;
    VT[0][(d40 + 2) * VSTRIDE + key0] = v0.z;
    VT[0][(d40 + 3) * VSTRIDE + key0] = v0.w;
    VT[0][(d40 + 0) * VSTRIDE + key1] = v1.x;
    VT[0][(d40 + 1) * VSTRIDE + key1] = v1.y;
    VT[0][(d40 + 2) * VSTRIDE + key1] = v1.z;
    VT[0][(d40 + 3) * VSTRIDE + key1] = v1.w;
  }
  __syncthreads();

  for (int i = 0; i < nb; ++i) {
    const int p  = i & 1;
    const int n0 = start + (i << 4);

    /* issue next block's global loads; wait deferred to the LDS stores below */
    int j0 = n0 + 16 + key0, j1 = n0 + 16 + key1;
    if (j0 > kv_len - 1) j0 = kv_len - 1;
    if (j1 > kv_len - 1) j1 = kv_len - 1;
    float4 nk0 = *(const float4*)kv_src(k, k_cache, b, kvh, j0, cache_len, d40);
    float4 nk1 = *(const float4*)kv_src(k, k_cache, b, kvh, j1, cache_len, d40);
    float4 nv0 = *(const float4*)kv_src(v, v_cache, b, kvh, j0, cache_len, d40);
    float4 nv1 = *(const float4*)kv_src(v, v_cache, b, kvh, j1, cache_len, d40);

    /* ---- S = Q.K^T : group 4 fragment loads per wait ---- */
    const float* kb = &Ks[p][lo * KSTRIDE + 2 * hi];
    v8f s = {0.f, 0.f, 0.f, 0.f, 0.f, 0.f, 0.f, 0.f};
#pragma unroll
    for (int c = 0; c < 32; c += 4) {
      v2f b0 = *(const v2f*)(kb + 4 * c);
      v2f b1 = *(const v2f*)(kb + 4 * (c + 1));
      v2f b2 = *(const v2f*)(kb + 4 * (c + 2));
      v2f b3 = *(const v2f*)(kb + 4 * (c + 3));
      s = __builtin_amdgcn_wmma_f32_16x16x4_f32(false, qf[c],     false, b0, (short)0, s, false, false);
      s = __builtin_amdgcn_wmma_f32_16x16x4_f32(false, qf[c + 1], false, b1, (short)0, s, false, false);
      s = __builtin_amdgcn_wmma_f32_16x16x4_f32(false, qf[c + 2], false, b2, (short)0, s, false, false);
      s = __builtin_amdgcn_wmma_f32_16x16x4_f32(false, qf[c + 3], false, b3, (short)0, s, false, false);
    }

    const bool tail = (n0 + 15 > cache_len);
#pragma unroll
    for (int e = 0; e < 8; ++e) {
      float sv = s[e];
      if (tail) {
        int qlr = (wave * 16 + e + 8 * hi) & (QL_ - 1);
        if (n0 + lo > cache_len + qlr) sv = NEG_BIG;   /* causal + padding mask */
      }
      float rm = sv;
      rm = fmaxf(rm, __shfl_xor(rm, 8, 32));
      rm = fmaxf(rm, __shfl_xor(rm, 4, 32));
      rm = fmaxf(rm, __shfl_xor(rm, 2, 32));
      rm = fmaxf(rm, __shfl_xor(rm, 1, 32));
      float mnew = fmaxf(mrow[e], rm);
      float pv = __expf(sv - mnew);
      float rs = pv;
      rs += __shfl_xor(rs, 8, 32);
      rs += __shfl_xor(rs, 4, 32);
      rs += __shfl_xor(rs, 2, 32);
      rs += __shfl_xor(rs, 1, 32);
      float corr = __expf(mrow[e] - mnew);
      lrow[e] = lrow[e] * corr + rs;
      mrow[e] = mnew;
#pragma unroll
      for (int dt = 0; dt < 8; ++dt) acc[dt][e] *= corr;
      Ps[(wave * 16 + e + 8 * hi) * PSTRIDE + lo] = pv;
    }

    /* ---- O += P.V : two V-fragments per wait ---- */
    const float* vb = &VT[p][lo * VSTRIDE + 2 * hi];
#pragma unroll
    for (int kk = 0; kk < 4; ++kk) {
      v2f pf = *(const v2f*)&Ps[(wave * 16 + lo) * PSTRIDE + 4 * kk + 2 * hi];
#pragma unroll
      for (int dt = 0; dt < 8; dt += 2) {
        v2f v0 = *(const v2f*)(vb + (dt + 0) * 16 * VSTRIDE + 4 * kk);
        v2f v1 = *(const v2f*)(vb + (dt + 1) * 16 * VSTRIDE + 4 * kk);
        acc[dt]     = __builtin_amdgcn_wmma_f32_16x16x4_f32(false, pf, false, v0,
                                                            (short)0, acc[dt],     false, false);
        acc[dt + 1] = __builtin_amdgcn_wmma_f32_16x16x4_f32(false, pf, false, v1,
                                                            (short)0, acc[dt + 1], false, false);
      }
    }

    /* ---- stage next block into the other buffer ---- */
    {
      const int np = p ^ 1;
      *(float4*)&Ks[np][key0 * KSTRIDE + d40] = nk0;
      *(float4*)&Ks[np][key1 * KSTRIDE + d40] = nk1;
      VT[np][(d40 + 0) * VSTRIDE + key0] = nv0.x;
      VT[np][(d40 + 1) * VSTRIDE + key0] = nv0.y;
      VT[np][(d40 + 2) * VSTRIDE + key0] = nv0.z;
      VT[np][(d40 + 3) * VSTRIDE + key0] = nv0.w;
      VT[np][(d40 + 0) * VSTRIDE + key1] = nv1.x;
      VT[np][(d40 + 1) * VSTRIDE + key1] = nv1.y;
      VT[np][(d40 + 2) * VSTRIDE + key1] = nv1.z;
      VT[np][(d40 + 3) * VSTRIDE + key1] = nv1.w;
    }
    __syncthreads();
  }

  if (SPLIT == 1) {
    /* ---- single-pass: normalize and store ---- */
#pragma unroll
    for (int e = 0; e < 8; ++e) {
      float inv = 1.0f / lrow[e];
      int rr  = wave * 16 + e + 8 * hi;
      int qlo = rr & (QL_ - 1);
      int ho  = kvh * REP_ + (rr >> 5);
      float* orow = out + (((size_t)b * QL_ + qlo) * H_ + ho) * (size_t)D_;
#pragma unroll
      for (int dt = 0; dt < 8; ++dt)
        orow[dt * 16 + lo] = acc[dt][e] * inv;
    }
  } else {
    /* ---- split-K: store unnormalized partials + (m, l) per row ---- */
    float* base = ws + (((size_t)(b * HKV_ + kvh)) * SPLIT + sid) * WSROW;
#pragma unroll
    for (int e = 0; e < 8; ++e) {
      int rr = wave * 16 + e + 8 * hi;
#pragma unroll
      for (int dt = 0; dt < 8; ++dt)
        base[(size_t)rr * D_ + dt * 16 + lo] = acc[dt][e];
      if (lo == 0) {
        base[(size_t)NROWS * D_ + rr]         = mrow[e];
        base[(size_t)NROWS * D_ + NROWS + rr] = lrow[e];
      }
    }
  }
}

/* log-sum-exp merge of split-K partials */
template <int SPLIT>
__global__ __launch_bounds__(256, 1)
void fa_combine(const float* __restrict__ ws, float* __restrict__ out) {
  const int bk  = blockIdx.x;              /* b*HKV + kvh */
  const int b   = bk / HKV_;
  const int kvh = bk % HKV_;
  const int tid = threadIdx.x;
  const int row = tid >> 1;                /* 0..127 */
  const int d0  = (tid & 1) * 64;

  const float* base0 = ws + (size_t)bk * SPLIT * WSROW;
  float m = NEG_BIG;
#pragma unroll
  for (int s = 0; s < SPLIT; ++s)
    m = fmaxf(m, base0[s * WSROW + (size_t)NROWS * D_ + row]);
  float sc[SPLIT];
  float lsum = 0.f;
#pragma unroll
  for (int s = 0; s < SPLIT; ++s) {
    float ms = base0[s * WSROW + (size_t)NROWS * D_ + row];
    float c  = __expf(ms - m);
    sc[s] = c;
    lsum += base0[s * WSROW + (size_t)NROWS * D_ + NROWS + row] * c;
  }
  const float inv = 1.0f / lsum;

  const int qlo = row & (QL_ - 1);
  const int ho  = kvh * REP_ + (row >> 5);
  float* orow = out + (((size_t)b * QL_ + qlo) * H_ + ho) * (size_t)D_;
#pragma unroll
  for (int dd = 0; dd < 64; dd += 4) {
    float4 o = {0.f, 0.f, 0.f, 0.f};
#pragma unroll
    for (int s = 0; s < SPLIT; ++s) {
      float4 a = *(const float4*)&base0[s * WSROW + (size_t)row * D_ + d0 + dd];
      o.x += a.x * sc[s]; o.y += a.y * sc[s];
      o.z += a.z * sc[s]; o.w += a.w * sc[s];
    }
    o.x *= inv; o.y *= inv; o.z *= inv; o.w *= inv;
    *(float4*)&orow[d0 + dd] = o;
  }
}

extern "C" void kernel_launch(void* const* d_in, const int* in_sizes, int n_in,
                              void* d_out, int out_size, void* d_ws, size_t ws_size,
                              hipStream_t stream) {
  (void)in_sizes; (void)n_in; (void)out_size;
  const float* q  = (const float*)d_in[0];
  const float* k  = (const float*)d_in[1];
  const float* v  = (const float*)d_in[2];
  const float* kc = (const float*)d_in[3];
  const float* vc = (const float*)d_in[4];
  const int*   cl = (const int*)d_in[5];
  float* out = (float*)d_out;
  float* ws  = (float*)d_ws;

  constexpr int SPLIT = 4;
  const size_t need = (size_t)B_ * HKV_ * SPLIT * WSROW * sizeof(float);
  if (ws != nullptr && ws_size >= need) {
    fa_kvcache_f32_wmma<SPLIT><<<dim3(B_ * HKV_ * SPLIT), dim3(256), 0, stream>>>(
        q, k, v, kc, vc, cl, out, ws);
    fa_combine<SPLIT><<<dim3(B_ * HKV_), dim3(256), 0, stream>>>(ws, out);
  } else {
    fa_kvcache_f32_wmma<1><<<dim3(B_ * HKV_), dim3(256), 0, stream>>>(
        q, k, v, kc, vc, cl, out, nullptr);
  }
}